// Interpolation_16028817949313
// MI455X (gfx1250) — compile-verified
//
#include <hip/hip_runtime.h>
#include <hip/hip_bf16.h>
#include <stdint.h>

#define GRID_MAX 63
#define IMG_FLOATS (64 * 64 * 3)        // 12288 floats = 48 KB
#define IMG_VEC4   (IMG_FLOATS / 4)     // 3072 float4 chunks

typedef float v4f __attribute__((ext_vector_type(4)));

#if defined(__AMDGCN__)
// Builtin wants pointers to GCC-style int4 vectors (per hipcc diagnostic).
typedef int v4i_gcc __attribute__((vector_size(16)));
typedef __attribute__((address_space(1))) v4i_gcc* g_v4i_p;   // global (printed as __device__)
typedef __attribute__((address_space(3))) v4i_gcc* l_v4i_p;   // LDS
#endif

__device__ __forceinline__ void async_copy_b128(const void* gsrc, void* lds_dst) {
#if defined(__AMDGCN__) && __has_builtin(__builtin_amdgcn_global_load_async_to_lds_b128)
  // Generic LDS pointer: low 32 bits are the LDS byte offset (aperture rule).
  __builtin_amdgcn_global_load_async_to_lds_b128(
      (g_v4i_p)(uintptr_t)gsrc,
      (l_v4i_p)(uint32_t)(uintptr_t)lds_dst,
      /*imm offset*/ 0, /*cpol*/ 0);
#else
  uint32_t lds_off = (uint32_t)(uintptr_t)lds_dst;
  uint64_t gaddr   = (uint64_t)(uintptr_t)gsrc;
  asm volatile("global_load_async_to_lds_b128 %0, %1, off"
               :
               : "v"(lds_off), "v"(gaddr)
               : "memory");
#endif
}

__device__ __forceinline__ void wait_async_zero() {
#if defined(__AMDGCN__) && __has_builtin(__builtin_amdgcn_s_wait_asynccnt)
  __builtin_amdgcn_s_wait_asynccnt(0);
#else
  asm volatile("s_wait_asynccnt 0x0" ::: "memory");
#endif
}

__global__ __launch_bounds__(256) void
Interpolation_16028817949313_kernel(const float* __restrict__ x,
                                    const float* __restrict__ image,
                                    float* __restrict__ out, int n) {
  __shared__ float simg[IMG_FLOATS];

  // ---- Stage the 48 KB image into LDS with CDNA5 async global->LDS DMA ----
  {
    const v4f* img4 = (const v4f*)image;
    v4f* s4 = (v4f*)simg;
    for (int i = threadIdx.x; i < IMG_VEC4; i += blockDim.x) {
      async_copy_b128((const void*)(img4 + i), (void*)(s4 + i));
    }
    wait_async_zero();   // this wave's async ops done
    __syncthreads();     // all waves' staging visible block-wide
  }

  const int tid    = blockIdx.x * blockDim.x + threadIdx.x;
  const int stride = gridDim.x * blockDim.x;
  const int nquad  = n >> 2;

  const v4f* x4  = (const v4f*)x;   // 2 points per float4
  v4f*      out4 = (v4f*)out;       // 4/3 points per float4

  for (int q = tid; q < nquad; q += stride) {
    v4f a = __builtin_nontemporal_load(x4 + 2 * q + 0);
    v4f b = __builtin_nontemporal_load(x4 + 2 * q + 1);

    float px[4] = {a.x, a.z, b.x, b.z};
    float py[4] = {a.y, a.w, b.y, b.w};
    float r[12];

#pragma unroll
    for (int k = 0; k < 4; ++k) {
      int lx = __float2int_rd(px[k]);
      int ly = __float2int_rd(py[k]);
      // (high - low) == 1.0f exactly for 0 <= low <= 62 -> division elided, bit-exact
      float wx = (float)(lx + 1) - px[k];
      float wy = (float)(ly + 1) - py[k];
      float w  = wx * wy;
      int ix = (lx < GRID_MAX) ? lx : GRID_MAX;
      int iy = (ly < GRID_MAX) ? ly : GRID_MAX;
      int base = (ix * 64 + iy) * 3;
      r[3 * k + 0] = simg[base + 0] * w;
      r[3 * k + 1] = simg[base + 1] * w;
      r[3 * k + 2] = simg[base + 2] * w;
    }

    v4f o0 = {r[0], r[1], r[2],  r[3]};
    v4f o1 = {r[4], r[5], r[6],  r[7]};
    v4f o2 = {r[8], r[9], r[10], r[11]};
    __builtin_nontemporal_store(o0, out4 + 3 * q + 0);
    __builtin_nontemporal_store(o1, out4 + 3 * q + 1);
    __builtin_nontemporal_store(o2, out4 + 3 * q + 2);
  }

  // Tail (n % 4 points) — scalar path
  const int rem = n & 3;
  if (tid < rem) {
    int i = (nquad << 2) + tid;
    float pxx = x[2 * i + 0];
    float pyy = x[2 * i + 1];
    int lx = __float2int_rd(pxx);
    int ly = __float2int_rd(pyy);
    float w = ((float)(lx + 1) - pxx) * ((float)(ly + 1) - pyy);
    int ix = (lx < GRID_MAX) ? lx : GRID_MAX;
    int iy = (ly < GRID_MAX) ? ly : GRID_MAX;
    int base = (ix * 64 + iy) * 3;
    out[3 * i + 0] = simg[base + 0] * w;
    out[3 * i + 1] = simg[base + 1] * w;
    out[3 * i + 2] = simg[base + 2] * w;
  }
}

extern "C" void kernel_launch(void* const* d_in, const int* in_sizes, int n_in,
                              void* d_out, int out_size, void* d_ws, size_t ws_size,
                              hipStream_t stream) {
  const float* x     = (const float*)d_in[0];   // (N, 2) float32
  const float* image = (const float*)d_in[1];   // (64, 64, 3) float32
  float* out         = (float*)d_out;           // (N, 3) float32

  int n = in_sizes[0] / 2;
  int nquad = n >> 2;

  const int threads = 256;
  long long want = ((long long)nquad + threads - 1) / threads;
  int blocks = (int)((want < 1) ? 1 : (want > 2048 ? 2048 : want));

  Interpolation_16028817949313_kernel<<<blocks, threads, 0, stream>>>(x, image, out, n);
}